// kernel_generated_1_21887153341257
// MI455X (gfx1250) — compile-verified
//
#include <hip/hip_runtime.h>

typedef __attribute__((ext_vector_type(2))) float v2f;
typedef __attribute__((ext_vector_type(8))) float v8f;

#define NCH   256
#define KTOT  2304            // 9 * 256 : K ordered as (i*3+k)*256 + cin
#define KG    (KTOT / 4)      // 576 K-groups of 4
#define HW    784             // 28*28

// Kernel 1: Weff[oc][(i*3+k)*256 + cin] = w1[l,n,i,k] * w2[p,cg,k]
//   oc = l*8+p, cin = cg*4+n
__global__ void build_weff_k(const float* __restrict__ w1, const float* __restrict__ w2,
                             float* __restrict__ weff) {
    int tid = blockIdx.x * blockDim.x + threadIdx.x;
    if (tid >= NCH * KTOT) return;
    int oc  = tid / KTOT;
    int r   = tid % KTOT;
    int blk = r >> 8;        // i*3+k
    int cin = r & 255;
    int i = blk / 3, k = blk % 3;
    int l = oc >> 3, p = oc & 7;
    int cg = cin >> 2, n = cin & 3;
    float a = w1[((l * 4 + n) * 3 + i) * 3 + k];   // w1: (32,4,3,3)
    float b = w2[(p * 64 + cg) * 3 + k];           // w2: (8,64,3)
    weff[tid] = a * b;
}

// Kernel 2: packed im2col.  Bpack[kg][s][e] = patch value for K = kg*4+e at
// spatial s, with K decomposed as blk = K/256 (= i*3+k), cin = K%256.
// Zero-padding handled here.  Layout puts the 4 K-values of a group adjacent,
// so the GEMM does ONE b64 load per lane per WMMA for B.
__global__ void build_bpack_k(const float* __restrict__ x, float* __restrict__ bpack) {
    int tid = blockIdx.x * blockDim.x + threadIdx.x;
    if (tid >= KTOT * HW) return;          // 1,806,336 elements
    int e  = tid & 3;
    int t  = tid >> 2;                     // kg*784 + s
    int s  = t % HW;
    int kg = t / HW;
    int blk = kg >> 6;                     // kg / 64  -> i*3+k
    int cin = ((kg & 63) << 2) | e;        // (kg%64)*4 + e
    int i = blk / 3, k = blk % 3;
    int ho = s / 28, wo = s % 28;
    int row = ho + i - 1, col = wo + k - 1;
    float v = 0.0f;
    if (row >= 0 && row < 28 && col >= 0 && col < 28)
        v = x[cin * HW + row * 28 + col];
    bpack[tid] = v;
}

// Kernel 3: GEMM [256 x 2304] x [2304 x 784] via V_WMMA_F32_16X16X4_F32.
// One wave = one 16(oc) x 16(spatial) tile; 576 WMMA steps, each fed by
// exactly one b64 A load and one b64 B load per lane.
__global__ void conv_wmma_k(const float* __restrict__ weff, const float* __restrict__ bpack,
                            float* __restrict__ out) {
    int wave   = (blockIdx.x * blockDim.x + threadIdx.x) >> 5;
    int lane   = threadIdx.x & 31;
    int m_tile = wave & 15;   // 16 tiles over 256 out channels
    int s_tile = wave >> 4;   // 49 tiles over 784 spatial positions
    int lr     = lane & 15;
    int half   = lane >> 4;

    int s    = s_tile * 16 + lr;          // spatial column (N index)
    int rowA = m_tile * 16 + lr;          // A row = M index (same both halves)

    // A: half 0 holds K = {k0, k0+1}; half 1 holds {k0+2, k0+3}
    const float* ap = weff + rowA * KTOT + 2 * half;
    // B: same split, and the 4 K-values of a group are adjacent in Bpack
    const float* bp = bpack + s * 4 + 2 * half;

    v8f acc = {};
    #pragma unroll 8
    for (int kg = 0; kg < KG; ++kg) {
        v2f a = *(const v2f*)(ap + kg * 4);          // b64, stride 16 B
        v2f b = *(const v2f*)(bp + kg * (HW * 4));   // b64, stride 12544 B
        acc = __builtin_amdgcn_wmma_f32_16x16x4_f32(
            false, a, false, b, (short)0, acc, false, false);
    }

    // D layout: acc[r] -> M = r + 8*half, N = lr
    #pragma unroll
    for (int r = 0; r < 8; ++r) {
        int oc = m_tile * 16 + half * 8 + r;
        out[oc * HW + s] = acc[r];
    }
}

extern "C" void kernel_launch(void* const* d_in, const int* in_sizes, int n_in,
                              void* d_out, int out_size, void* d_ws, size_t ws_size,
                              hipStream_t stream) {
    const float* x  = (const float*)d_in[0];   // (1,256,28,28)
    const float* w1 = (const float*)d_in[1];   // (32,4,3,3)
    const float* w2 = (const float*)d_in[2];   // (8,64,3)
    float* out = (float*)d_out;                // (1,256,28,28)

    float* weff  = (float*)d_ws;               // 256*2304 floats  = 2.25 MB
    float* bpack = weff + NCH * KTOT;          // 2304*784 floats  = 7.22 MB

    build_weff_k<<<(NCH * KTOT + 255) / 256, 256, 0, stream>>>(w1, w2, weff);
    build_bpack_k<<<(KTOT * HW + 255) / 256, 256, 0, stream>>>(x, bpack);
    // 784 tiles, 4 waves (128 threads) per block -> 196 blocks
    conv_wmma_k<<<196, 128, 0, stream>>>(weff, bpack, out);
}